// ExponentialMovingAverage_79929341378572
// MI455X (gfx1250) — compile-verified
//
#include <hip/hip_runtime.h>
#include <cstdint>
#include <cstddef>

#ifndef __has_builtin
#define __has_builtin(x) 0
#endif

// TDM builtins are target-specific: only check/use them during the DEVICE pass.
// (The host pass of hipcc doesn't register amdgcn builtins -> __has_builtin=0.)
#if defined(__HIP_DEVICE_COMPILE__)
  #if !__has_builtin(__builtin_amdgcn_tensor_load_to_lds)
    #error "CDNA5 TDM builtin __builtin_amdgcn_tensor_load_to_lds not available on device toolchain"
  #endif
  #if !__has_builtin(__builtin_amdgcn_s_wait_tensorcnt)
    #error "CDNA5 builtin __builtin_amdgcn_s_wait_tensorcnt not available on device toolchain"
  #endif
  #define HAVE_TDM 1
#else
  #define HAVE_TDM 0   // host parse uses the plain-load fallback path
#endif

#define BB 16
#define TT 8000
#define CC 512
#define LL 16
#define GG 500          // TT / LL
#define XSTRIDE 260     // 16x16 dwords per b-slice + 4 pad dwords (TDM pad)

typedef float v2f __attribute__((ext_vector_type(2)));
typedef float v8f __attribute__((ext_vector_type(8)));
typedef unsigned int v4u __attribute__((ext_vector_type(4)));
typedef int v4i __attribute__((ext_vector_type(4)));
typedef int v8i __attribute__((ext_vector_type(8)));

__device__ __forceinline__ float clamp01(float v) { return fminf(fmaxf(v, 0.0f), 1.0f); }

// ---------------------------------------------------------------------------
// Pass 0: per-channel coefficient tables (one-time, negligible cost).
//   atab[c][j] = (j < 16) ? 0 : k_c * d_c^(j-16)      (zero-padded scan matrix)
//   dtab[c][m] = d_c^(m+1)                            (carry fix-up column)
// Zero-padding removes all compares/selects from the pass-3 inner loop.
// ---------------------------------------------------------------------------
__global__ __launch_bounds__(256) void ema_pass0_tables(
    const float* __restrict__ smooth, float* __restrict__ atab,
    float* __restrict__ dtab) {
  const int tid = blockIdx.x * blockDim.x + threadIdx.x;  // 0 .. CC*32-1
  const int c = tid >> 5;
  const int j = tid & 31;
  const float k = clamp01(smooth[c]);
  const float d = 1.0f - k;
  const float d2 = d * d, d4 = d2 * d2, d8 = d4 * d4;

  const int e = j - 16;
  const int ee = (e < 0) ? 0 : e;
  float p = k;
  if (ee & 1) p *= d;
  if (ee & 2) p *= d2;
  if (ee & 4) p *= d4;
  if (ee & 8) p *= d8;
  atab[c * 32 + j] = (e < 0) ? 0.0f : p;

  if (j < 16) {
    const int m = j + 1;                 // 1..16
    float q = 1.0f;
    if (m & 1) q *= d;
    if (m & 2) q *= d2;
    if (m & 4) q *= d4;
    if (m & 8) q *= d8;
    if (m & 16) q *= d8 * d8;
    dtab[c * 16 + j] = q;
  }
}

// ---------------------------------------------------------------------------
// Pass 1: per 16-step chunk, Horner-accumulate  h_g = sum_j k * d^(15-j) * x_j
// Thread = (g, b, c4) handling a float4 of channels. Fully coalesced stream.
// ---------------------------------------------------------------------------
__global__ __launch_bounds__(128) void ema_pass1_aggregate(
    const float* __restrict__ x, const float* __restrict__ smooth,
    float* __restrict__ hbuf) {
  const int gb = blockIdx.x;          // g*BB + b
  const int g = gb >> 4;
  const int b = gb & 15;
  const int c = threadIdx.x * 4;      // 0..508

  const float k0 = clamp01(smooth[c + 0]);
  const float k1 = clamp01(smooth[c + 1]);
  const float k2 = clamp01(smooth[c + 2]);
  const float k3 = clamp01(smooth[c + 3]);
  const float d0 = 1.0f - k0, d1 = 1.0f - k1, d2 = 1.0f - k2, d3 = 1.0f - k3;

  const float* xp = x + ((size_t)b * TT + (size_t)g * LL) * CC + c;
  float h0 = 0.f, h1 = 0.f, h2 = 0.f, h3 = 0.f;
#pragma unroll
  for (int j = 0; j < LL; ++j) {
    const float4 v = *(const float4*)(xp + (size_t)j * CC);
    h0 = fmaf(d0, h0, k0 * v.x);
    h1 = fmaf(d1, h1, k1 * v.y);
    h2 = fmaf(d2, h2, k2 * v.z);
    h3 = fmaf(d3, h3, k3 * v.w);
  }
  float4 r; r.x = h0; r.y = h1; r.z = h2; r.w = h3;
  *(float4*)(hbuf + ((size_t)g * BB + b) * CC + c) = r;
}

// ---------------------------------------------------------------------------
// Pass 2: cross-chunk carry scan. carry[0] = x[:,0,:] (the reference's initial
// state; k*x0 + (1-k)*x0 == x0 so the uniform recurrence holds from t=0).
// carry[g+1] = h[g] + d^16 * carry[g].   Layout: carry[g][b][c].
// ---------------------------------------------------------------------------
__global__ __launch_bounds__(256) void ema_pass2_carry(
    const float* __restrict__ x, const float* __restrict__ smooth,
    const float* __restrict__ hbuf, float* __restrict__ carry) {
  const int tid = blockIdx.x * blockDim.x + threadIdx.x;  // 0..2047
  const int b = tid >> 7;
  const int c = (tid & 127) * 4;

  float k0 = clamp01(smooth[c + 0]), k1 = clamp01(smooth[c + 1]);
  float k2 = clamp01(smooth[c + 2]), k3 = clamp01(smooth[c + 3]);
  float D0 = 1.0f - k0, D1 = 1.0f - k1, D2 = 1.0f - k2, D3 = 1.0f - k3;
#pragma unroll
  for (int s = 0; s < 4; ++s) { D0 *= D0; D1 *= D1; D2 *= D2; D3 *= D3; }

  const float4 x0 = *(const float4*)(x + (size_t)b * TT * CC + c);
  *(float4*)(carry + (size_t)b * CC + c) = x0;   // g = 0
  float a0 = x0.x, a1 = x0.y, a2 = x0.z, a3 = x0.w;

  for (int g = 0; g < GG - 1; ++g) {
    const float4 hv = *(const float4*)(hbuf + ((size_t)g * BB + b) * CC + c);
    a0 = fmaf(D0, a0, hv.x);
    a1 = fmaf(D1, a1, hv.y);
    a2 = fmaf(D2, a2, hv.z);
    a3 = fmaf(D3, a3, hv.w);
    float4 o; o.x = a0; o.y = a1; o.z = a2; o.w = a3;
    *(float4*)(carry + ((size_t)(g + 1) * BB + b) * CC + c) = o;
  }
}

// ---------------------------------------------------------------------------
// Pass 3: WMMA reconstruction. For each (channel-group of 16, t-chunk of 16):
//   z(16t x 16b)  = M_c(16x16) * X(16t x 16b)     via 4x V_WMMA_F32_16X16X4_F32
//   out           = z + dcol * carry_row          via a 5th WMMA (rank-1 update)
// The (16c x 16t x 16b) input tile is DMA'd into LDS by the Tensor Data Mover
// (3-D descriptor, 4-dword pad per 256-dword b-slice => 260-dword slice stride),
// outputs transposed back through LDS for coalesced 64B cacheline stores.
// ---------------------------------------------------------------------------
__global__ __launch_bounds__(128) void ema_pass3_wmma(
    const float* __restrict__ x, const float* __restrict__ atab,
    const float* __restrict__ dtab, const float* __restrict__ carry,
    float* __restrict__ out) {
  __shared__ __align__(16) float xt[16 * XSTRIDE];  // xt[b*260 + t*16 + c_local]
  __shared__ __align__(16) float ot[16 * XSTRIDE];  // same layout for output
  __shared__ float at[16 * 32];                     // atab rows for 16 channels
  __shared__ float dt[16 * 16];                     // dtab rows for 16 channels
  __shared__ float ct[256];                         // carry[b*16 + c_local]

  const int tid = threadIdx.x;
  const int g = blockIdx.x % GG;
  const int cg = blockIdx.x / GG;
  const int c0 = cg * 16;
  const size_t tileElem = (size_t)g * LL * CC + c0;  // (b=0, t=g*16, c0)

#if HAVE_TDM
  // ---- TDM: one descriptor per workgroup loads the whole 16x16x16 f32 tile.
  if (tid < 32) {  // TDM ignores EXEC; gate by wave so only one DMA is issued
    const unsigned ldsa = (unsigned)(size_t)(void*)xt;
    const unsigned long long ga = (unsigned long long)(const void*)(x + tileElem);
    v4u g0;
    g0[0] = 1u;                                   // count=1 (valid descriptor)
    g0[1] = ldsa;                                 // lds_addr
    g0[2] = (unsigned)ga;                         // global_addr[31:0]
    g0[3] = (unsigned)((ga >> 32) & 0x01FFFFFFu)  // global_addr[56:32]
            | 0x80000000u;                        // type=2 ("image")
    v8i g1;
    g1[0] = (int)0x07D20000u;  // data_size=4B | pad_en | pad_interval=256dw | pad=4dw
    g1[1] = (int)0x02000000u;  // tensor_dim0 = 512 (low16 << 16)
    g1[2] = (int)0x1F400000u;  // tensor_dim1 = 8000 (low16 << 16) | dim0 hi16 = 0
    g1[3] = (int)0x00100000u;  // tile_dim0 = 16 (<<16) | dim1 hi16 = 0
    g1[4] = (int)0x00100010u;  // tile_dim2 = 16 (<<16) | tile_dim1 = 16
    g1[5] = (int)CC;           // tensor_dim0_stride = 512   (t step)
    g1[6] = (int)0x80000000u;  // dim1_stride low16 (<<16) | dim0_stride hi16
    g1[7] = (int)0x0000003Eu;  // dim1_stride = T*C = 4,096,000 (>>16)  (b step)
    v4i g2; g2[0] = BB; g2[1] = 0; g2[2] = 0; g2[3] = 0;  // tensor_dim2 = 16
    v4i g3; g3[0] = 0; g3[1] = 0; g3[2] = 0; g3[3] = 0;
#if defined(__clang_major__) && (__clang_major__ >= 23)
    v8i g4 = (v8i)0;
    __builtin_amdgcn_tensor_load_to_lds(g0, g1, g2, g3, g4, 0);
#else
    __builtin_amdgcn_tensor_load_to_lds(g0, g1, g2, g3, 0);
#endif
  }
#endif  // HAVE_TDM

  // ---- Stage coefficient tables + carries (overlaps with the TDM transfer).
  {
    const int i0 = tid, i1 = tid + 128;
    ct[i0] = carry[((size_t)g * BB + (i0 >> 4)) * CC + c0 + (i0 & 15)];
    ct[i1] = carry[((size_t)g * BB + (i1 >> 4)) * CC + c0 + (i1 & 15)];
#pragma unroll
    for (int r = 0; r < 4; ++r)
      at[tid + r * 128] = atab[(size_t)c0 * 32 + tid + r * 128];
#pragma unroll
    for (int r = 0; r < 2; ++r)
      dt[tid + r * 128] = dtab[(size_t)c0 * 16 + tid + r * 128];
  }

#if HAVE_TDM
  if (tid < 32) __builtin_amdgcn_s_wait_tensorcnt(0);
  __syncthreads();
#else
  // Host-parse / non-TDM fallback: cooperative coalesced float4 staging.
  __syncthreads();
  for (int r = 0; r < 8; ++r) {
    const int flat = tid + r * 128;
    const int row = flat >> 2, q = flat & 3;
    const int bb = row >> 4, t = row & 15;
    const float4 v = *(const float4*)(x + tileElem + ((size_t)bb * TT + t) * CC + q * 4);
    *(float4*)&xt[bb * XSTRIDE + t * 16 + q * 4] = v;
  }
  __syncthreads();
#endif

  const int lane = tid & 31;
  const int w = tid >> 5;
  const int Mrow = lane & 15;     // A row M; B column N (= batch); carry's b
  const int hi = lane >> 4;
  const int aBase = Mrow - 2 * hi + 16;  // atab index of k*d^(M-K0) at kb=0

#pragma unroll
  for (int i = 0; i < 4; ++i) {
    const int cl = w * 4 + i;     // channel handled by this wave this iteration
    const float* atc = &at[cl * 32];
    v8f acc = {0.f, 0.f, 0.f, 0.f, 0.f, 0.f, 0.f, 0.f};
#pragma unroll
    for (int kb = 0; kb < 16; kb += 4) {
      v2f A, Bm;
      A.x = atc[aBase - kb];           // k*d^(M-K0), 0 if K0 > M
      A.y = atc[aBase - kb - 1];       // k*d^(M-K1), 0 if K1 > M
      const int K0 = kb + 2 * hi;
      Bm.x = xt[Mrow * XSTRIDE + K0 * 16 + cl];
      Bm.y = xt[Mrow * XSTRIDE + (K0 + 1) * 16 + cl];
      acc = __builtin_amdgcn_wmma_f32_16x16x4_f32(
          false, A, false, Bm, (short)0, acc, false, false);
    }
    // Carry fix-up as a rank-1 WMMA:  D[M][N] += d^(M+1) * carry[N]
    v2f Ap, Bp;
    Ap.x = hi ? 0.0f : dt[cl * 16 + Mrow];
    Ap.y = 0.0f;
    Bp.x = hi ? 0.0f : ct[Mrow * 16 + cl];
    Bp.y = 0.0f;
    acc = __builtin_amdgcn_wmma_f32_16x16x4_f32(
        false, Ap, false, Bp, (short)0, acc, false, false);

#pragma unroll
    for (int v = 0; v < 8; ++v)
      ot[Mrow * XSTRIDE + (v + hi * 8) * 16 + cl] = acc[v];
  }
  __syncthreads();

  // Transposed write-out: each (b,t) row is 16 channels = one 64B cacheline.
#pragma unroll
  for (int r = 0; r < 8; ++r) {
    const int flat = tid + r * 128;
    const int row = flat >> 2, q = flat & 3;
    const int bb = row >> 4, t = row & 15;
    *(float4*)(out + tileElem + ((size_t)bb * TT + t) * CC + q * 4) =
        *(const float4*)&ot[bb * XSTRIDE + t * 16 + q * 4];
  }
}

extern "C" void kernel_launch(void* const* d_in, const int* in_sizes, int n_in,
                              void* d_out, int out_size, void* d_ws, size_t ws_size,
                              hipStream_t stream) {
  (void)in_sizes; (void)n_in; (void)out_size; (void)ws_size;
  const float* x = (const float*)d_in[0];
  const float* smooth = (const float*)d_in[1];
  float* out = (float*)d_out;
  float* hbuf = (float*)d_ws;                       // GG*BB*CC f32 = 16 MB
  float* carry = hbuf + (size_t)GG * BB * CC;       // GG*BB*CC f32 = 16 MB
  float* atab = carry + (size_t)GG * BB * CC;       // CC*32 f32 = 64 KB
  float* dtab = atab + (size_t)CC * 32;             // CC*16 f32 = 32 KB

  ema_pass0_tables<<<(CC * 32) / 256, 256, 0, stream>>>(smooth, atab, dtab);
  ema_pass1_aggregate<<<GG * BB, CC / 4, 0, stream>>>(x, smooth, hbuf);
  ema_pass2_carry<<<(BB * CC / 4) / 256, 256, 0, stream>>>(x, smooth, hbuf, carry);
  ema_pass3_wmma<<<GG * (CC / 16), 128, 0, stream>>>(x, atab, dtab, carry, out);
}